// GCNEncoder_8916352106736
// MI455X (gfx1250) — compile-verified
//
#include <hip/hip_runtime.h>

#define NN 50000
#define NE 1600000
#define DD 128
#define NG 64
#define MT 80            // rows per block (50000 = 625 * 80), 5 x 16-row tiles per wave
#define NTILE 5
#define LSTRIDE 132      // LDS row stride (floats): conflict-free column reads

typedef __attribute__((ext_vector_type(2))) float v2f;
typedef __attribute__((ext_vector_type(8))) float v8f;

// ---------------- utility kernels ----------------

__global__ void k_fill(float* __restrict__ p, float v, int n) {
    int i = blockIdx.x * blockDim.x + threadIdx.x;
    if (i < n) p[i] = v;
}

// deg[dst] += ew  (deg pre-initialized to 1.0 for the self-loop weight)
__global__ void k_deg(const int* __restrict__ dst, const float* __restrict__ ew,
                      float* __restrict__ deg) {
    int e = blockIdx.x * blockDim.x + threadIdx.x;
    if (e < NE) atomicAdd(&deg[dst[e]], ew[e]);
}

// in-place deg -> dinv = rsqrt(deg)
__global__ void k_dinv(float* __restrict__ d) {
    int i = blockIdx.x * blockDim.x + threadIdx.x;
    if (i < NN) { float v = d[i]; d[i] = (v > 0.f) ? rsqrtf(v) : 0.f; }
}

// norm[e] = dinv[src] * ew * dinv[dst]   (reused across all 3 layers)
__global__ void k_norm(const int* __restrict__ src, const int* __restrict__ dst,
                       const float* __restrict__ ew, const float* __restrict__ dinv,
                       float* __restrict__ norm) {
    int e = blockIdx.x * blockDim.x + threadIdx.x;
    if (e < NE) norm[e] = dinv[src[e]] * ew[e] * dinv[dst[e]];
}

// ---------------- WMMA GEMM: xw = relu?(hin) @ W ; out = bias + dinv^2 * xw ----------------
// grid.x = NN/MT = 625, block = 256 threads = 8 wave32.
// Block computes rows [MT*bid, MT*bid+80) x all 128 cols; wave w owns cols [16w,16w+16).
// Each wave keeps NTILE=5 independent 16x16 accumulator chains sharing one B fragment
// per K-step (ILP across the WMMA pipe + 5x B-load amortization).
// V_WMMA_F32_16X16X4_F32 layouts (ISA 7.12.2):
//   A (16x4): lane(0-15) -> M=lane, vgpr r -> K = 2*half + r
//   B (4x16): lane(0-15) -> N=lane, vgpr r -> K = 2*half + r
//   C/D     : lane(0-15) -> N=lane, vgpr r -> M = r + 8*half
__global__ __launch_bounds__(256)
void k_gemm_wmma(const float* __restrict__ hin, const float* __restrict__ W,
                 const float* __restrict__ bias, const float* __restrict__ dinv,
                 float* __restrict__ xw, float* __restrict__ out, int relu_in)
{
    __shared__ float As[MT * LSTRIDE];        // 80*132*4 = 42.2 KB
    const int tid  = threadIdx.x;
    const int wave = tid >> 5;
    const int lane = tid & 31;
    const int half = lane >> 4;
    const int lid  = lane & 15;
    const int r0   = blockIdx.x * MT;

    // stage A tile (80x128) into LDS via b128 loads, optional ReLU on load
    #pragma unroll
    for (int i = 0; i < (MT * DD) / (256 * 4); ++i) {   // 10 iterations
        int f  = (tid + i * 256) * 4;
        int rr = f >> 7, cc = f & 127;
        float4 v = *(const float4*)(hin + (size_t)(r0 + rr) * DD + cc);
        if (relu_in) {
            v.x = fmaxf(v.x, 0.f); v.y = fmaxf(v.y, 0.f);
            v.z = fmaxf(v.z, 0.f); v.w = fmaxf(v.w, 0.f);
        }
        *(float4*)(&As[rr * LSTRIDE + cc]) = v;
    }
    __syncthreads();

    const int c0 = wave * 16;
    v8f acc[NTILE];
    #pragma unroll
    for (int t = 0; t < NTILE; ++t) acc[t] = (v8f){};

    #pragma unroll
    for (int k = 0; k < DD; k += 4) {
        v2f b;
        b.x = W[(k + 2 * half + 0) * DD + c0 + lid];
        b.y = W[(k + 2 * half + 1) * DD + c0 + lid];
        #pragma unroll
        for (int t = 0; t < NTILE; ++t) {
            v2f a;
            a.x = As[(t * 16 + lid) * LSTRIDE + k + 2 * half + 0];
            a.y = As[(t * 16 + lid) * LSTRIDE + k + 2 * half + 1];
            acc[t] = __builtin_amdgcn_wmma_f32_16x16x4_f32(
                false, a, false, b, (short)0, acc[t], false, false);
        }
    }

    // epilogue: store xw; out = bias + dinv^2 * xw (self-loop term fused, full overwrite)
    const float bc = bias[c0 + lid];
    #pragma unroll
    for (int t = 0; t < NTILE; ++t) {
        #pragma unroll
        for (int r = 0; r < 8; ++r) {
            int row  = r0 + t * 16 + r + 8 * half;
            float di = dinv[row];
            float v  = acc[t][r];
            xw[(size_t)row * DD + c0 + lid]  = v;
            out[(size_t)row * DD + c0 + lid] = bc + di * di * v;
        }
    }
}

// ---------------- edge scatter: out[dst] += norm[e] * xw[src] ----------------
// one wave per edge; lane loads float4 (b128, 512B/wave coalesced gather), 4 L2 atomics
__global__ __launch_bounds__(256)
void k_scatter(const int* __restrict__ src, const int* __restrict__ dst,
               const float* __restrict__ norm, const float* __restrict__ xw,
               float* __restrict__ out)
{
    int e = blockIdx.x * 8 + (threadIdx.x >> 5);
    if (e >= NE) return;
    int lane = threadIdx.x & 31;
    int s = src[e], d = dst[e];
    float w = norm[e];
    float4 v = ((const float4*)(xw + (size_t)s * DD))[lane];
    float* q = out + (size_t)d * DD + lane * 4;
    atomicAdd(q + 0, w * v.x);
    atomicAdd(q + 1, w * v.y);
    atomicAdd(q + 2, w * v.z);
    atomicAdd(q + 3, w * v.w);
}

// ---------------- global mean pool ----------------
__global__ __launch_bounds__(256)
void k_pool(const float* __restrict__ h, const int* __restrict__ batch,
            float* __restrict__ sums, float* __restrict__ cnts)
{
    int node = blockIdx.x * 8 + (threadIdx.x >> 5);
    if (node >= NN) return;
    int lane = threadIdx.x & 31;
    int g = batch[node];
    float4 v = ((const float4*)(h + (size_t)node * DD))[lane];
    float* q = sums + g * DD + lane * 4;
    atomicAdd(q + 0, v.x);
    atomicAdd(q + 1, v.y);
    atomicAdd(q + 2, v.z);
    atomicAdd(q + 3, v.w);
    if (lane == 0) atomicAdd(&cnts[g], 1.0f);
}

__global__ void k_final(const float* __restrict__ sums, const float* __restrict__ cnts,
                        float* __restrict__ out) {
    int i = blockIdx.x * blockDim.x + threadIdx.x;
    if (i < NG * DD) out[i] = sums[i] / fmaxf(cnts[i / DD], 1.0f);
}

// ---------------- launch ----------------
extern "C" void kernel_launch(void* const* d_in, const int* in_sizes, int n_in,
                              void* d_out, int out_size, void* d_ws, size_t ws_size,
                              hipStream_t stream) {
    const float* x     = (const float*)d_in[0];
    const int*   ei    = (const int*)d_in[1];     // [2, NE]
    const float* ew    = (const float*)d_in[2];
    const int*   batch = (const int*)d_in[3];
    const float* Wm[3] = {(const float*)d_in[4], (const float*)d_in[6], (const float*)d_in[8]};
    const float* Bv[3] = {(const float*)d_in[5], (const float*)d_in[7], (const float*)d_in[9]};
    const int* src = ei;
    const int* dst = ei + NE;
    float* outp = (float*)d_out;

    // workspace layout (floats)
    float* ws   = (float*)d_ws;
    float* dinv = ws;                    // 50048 (padded)
    float* norm = dinv + 50048;          // NE
    float* xw   = norm + NE;             // NN*DD
    float* bufA = xw   + (size_t)NN * DD;
    float* bufB = bufA + (size_t)NN * DD;
    float* sums = bufB + (size_t)NN * DD; // NG*DD
    float* cnts = sums + NG * DD;         // NG (contiguous after sums)

    dim3 b256(256);

    // normalization precompute
    k_fill<<<(NN + 255) / 256, b256, 0, stream>>>(dinv, 1.0f, NN);       // self-loop weight
    k_deg <<<(NE + 255) / 256, b256, 0, stream>>>(dst, ew, dinv);
    k_dinv<<<(NN + 255) / 256, b256, 0, stream>>>(dinv);
    k_norm<<<(NE + 255) / 256, b256, 0, stream>>>(src, dst, ew, dinv, norm);

    // 3 GCN layers; relu fused into next layer's GEMM A-load
    const float* hin = x;
    float* outs[3] = {bufA, bufB, bufA};
    for (int l = 0; l < 3; ++l) {
        k_gemm_wmma<<<NN / MT, b256, 0, stream>>>(hin, Wm[l], Bv[l], dinv, xw, outs[l], l > 0);
        k_scatter  <<<(NE + 7) / 8, b256, 0, stream>>>(src, dst, norm, xw, outs[l]);
        hin = outs[l];
    }

    // global mean pool
    k_fill <<<(NG * DD + NG + 255) / 256, b256, 0, stream>>>(sums, 0.0f, NG * DD + NG);
    k_pool <<<(NN + 7) / 8, b256, 0, stream>>>(bufA, batch, sums, cnts);
    k_final<<<(NG * DD + 255) / 256, b256, 0, stream>>>(sums, cnts, outp);
}